// HashingLayer_74801150427836
// MI455X (gfx1250) — compile-verified
//
#include <hip/hip_runtime.h>

// Feature-hashing scatter-reduce for MI455X (gfx1250).
//   out[b, mask[i]] += signs[i] * x[b, i]
// Roofline: 512 MiB x streamed once + 16 MiB out + L2-resident 4 MiB
// mask/signs => pure bandwidth problem (~23 us at 23.3 TB/s). Compute is
// ~134M FMAs, trivial. CDNA5 path used: GLOBAL_LOAD_ASYNC_TO_LDS_B128
// pipeline (ASYNCcnt) + ds_add_f32 LDS histogram privatization.

#define IN_F   524288
#define OUT_F  16384
#define TPB    1024
#define VEC    4
#define DEPTH  4                     // pipeline depth (tiles in flight)
#define SPLIT  2                     // feature splits per batch row
#define TILE   (TPB * VEC)           // 4096 features per tile per block
#define FPB    (IN_F / SPLIT)        // 262144 features per block
#define NTILES (FPB / TILE)          // 64 tiles

static_assert(DEPTH == 4, "epilogue waits are hardcoded for DEPTH==4");
static_assert(NTILES > DEPTH, "pipeline needs more tiles than depth");

typedef float f32x4 __attribute__((ext_vector_type(4)));
typedef int   i32x4 __attribute__((ext_vector_type(4)));

#define GAS __attribute__((address_space(1)))
#define LAS __attribute__((address_space(3)))

// cpol: TH[2:0] | SCOPE[4:3];  TH_RT=0, TH_NT=1 (stream-once data)
template <int CPOL>
__device__ __forceinline__ void async_ld_b128(void* lds_dst, const void* gsrc) {
  __builtin_amdgcn_global_load_async_to_lds_b128(
      (GAS i32x4*)gsrc, (LAS i32x4*)lds_dst, /*ioffset=*/0, CPOL);
}

__global__ void __launch_bounds__(TPB, 1)
hash_scatter_kernel(const float* __restrict__ x,
                    const float* __restrict__ signs,
                    const int*   __restrict__ mask,
                    float* __restrict__ out)
{
  __shared__ float hist[OUT_F];          //  64 KiB private output row
  __shared__ float xs[DEPTH][TILE];      //  64 KiB x staging
  __shared__ float ss[DEPTH][TILE];      //  64 KiB signs staging
  __shared__ int   ms[DEPTH][TILE];      //  64 KiB mask staging (256 KiB total)

  const int t  = threadIdx.x;
  const int b  = blockIdx.x;             // batch row
  const int sp = blockIdx.y;             // feature split

  // zero the private histogram (vectorized: ds_store_b128)
  {
    const f32x4 z = {0.0f, 0.0f, 0.0f, 0.0f};
#pragma unroll
    for (int i = 0; i < OUT_F / (TPB * VEC); ++i)
      *(f32x4*)&hist[i * TPB * VEC + t * VEC] = z;
  }
  __syncthreads();

  const size_t fbase = (size_t)sp * FPB;
  const float* xrow = x + (size_t)b * IN_F + fbase;
  const float* srow = signs + fbase;
  const int*   mrow = mask + fbase;

  auto issue = [&](int k) {
    const int buf = k & (DEPTH - 1);
    const int gi  = k * TILE + t * VEC;
    async_ld_b128<1>(&xs[buf][t * VEC], xrow + gi);   // NT: streamed once
    async_ld_b128<0>(&ss[buf][t * VEC], srow + gi);   // RT: L2-hot
    async_ld_b128<0>(&ms[buf][t * VEC], mrow + gi);   // RT: L2-hot
  };

  auto load_tile = [&](int k, f32x4& xv, f32x4& sv, i32x4& mv) {
    const int buf = k & (DEPTH - 1);
    // each lane reads back exactly the 16B chunks it staged -> no barrier
    xv = *(const f32x4*)&xs[buf][t * VEC];
    sv = *(const f32x4*)&ss[buf][t * VEC];
    mv = *(const i32x4*)&ms[buf][t * VEC];
  };

  auto accum = [&](f32x4 xv, f32x4 sv, i32x4 mv) {
#pragma unroll
    for (int j = 0; j < VEC; ++j)
      __hip_atomic_fetch_add(&hist[mv[j]], xv[j] * sv[j],
                             __ATOMIC_RELAXED, __HIP_MEMORY_SCOPE_WORKGROUP);
  };

  // prologue: fill the pipeline
#pragma unroll
  for (int k = 0; k < DEPTH; ++k) issue(k);

  // steady state: tiles 0 .. NTILES-DEPTH (wait leaves 3 younger tiles in flight)
  for (int k = 0; k + (DEPTH - 1) < NTILES; ++k) {
    __builtin_amdgcn_s_wait_asynccnt(3 * (DEPTH - 1));
    f32x4 xv, sv; i32x4 mv;
    load_tile(k, xv, sv, mv);
    if (k + DEPTH < NTILES) {
      // ensure our ds_loads of this buffer retired before its async reuse
      asm volatile("s_wait_dscnt 0x0" ::: "memory");
      issue(k + DEPTH);
    }
    accum(xv, sv, mv);
  }

  // epilogue: drain the last DEPTH-1 tiles with tightening waits
  {
    f32x4 xv, sv; i32x4 mv;
    __builtin_amdgcn_s_wait_asynccnt(6);
    load_tile(NTILES - 3, xv, sv, mv);  accum(xv, sv, mv);
    __builtin_amdgcn_s_wait_asynccnt(3);
    load_tile(NTILES - 2, xv, sv, mv);  accum(xv, sv, mv);
    __builtin_amdgcn_s_wait_asynccnt(0);
    load_tile(NTILES - 1, xv, sv, mv);  accum(xv, sv, mv);
  }

  __syncthreads();

  // flush private histogram: SPLIT blocks share each output row -> atomic add
  float* orow = out + (size_t)b * OUT_F;
#pragma unroll
  for (int i = 0; i < OUT_F / (TPB * VEC); ++i) {
    const int o = i * TPB * VEC + t * VEC;
    f32x4 v = *(const f32x4*)&hist[o];
#pragma unroll
    for (int j = 0; j < VEC; ++j)
      __hip_atomic_fetch_add(&orow[o + j], v[j],
                             __ATOMIC_RELAXED, __HIP_MEMORY_SCOPE_AGENT);
  }
}

__global__ void zero_f32x4_kernel(float* __restrict__ p, int n4) {
  int i = blockIdx.x * blockDim.x + threadIdx.x;
  if (i < n4) {
    f32x4 z = {0.0f, 0.0f, 0.0f, 0.0f};
    *((f32x4*)p + i) = z;
  }
}

extern "C" void kernel_launch(void* const* d_in, const int* in_sizes, int n_in,
                              void* d_out, int out_size, void* d_ws, size_t ws_size,
                              hipStream_t stream) {
  (void)n_in; (void)d_ws; (void)ws_size;
  const float* x     = (const float*)d_in[0];
  const float* signs = (const float*)d_in[1];
  const int*   mask  = (const int*)d_in[2];
  float*       out   = (float*)d_out;

  const int B = in_sizes[0] / IN_F;      // 256

  // zero output (harness poisons d_out; we flush with atomic adds)
  const int n4 = (B * OUT_F) / 4;
  hipLaunchKernelGGL(zero_f32x4_kernel, dim3((n4 + 255) / 256), dim3(256), 0,
                     stream, out, n4);

  dim3 grid(B, SPLIT);
  hipLaunchKernelGGL(hash_scatter_kernel, grid, dim3(TPB), 0, stream,
                     x, signs, mask, out);
  (void)out_size;
}